// CUEQ_Encoder_14053132993058
// MI455X (gfx1250) — compile-verified
//
#include <hip/hip_runtime.h>
#include <hip/hip_bf16.h>

// ---------------------------------------------------------------------------
// CDNA5 (gfx1250) equivariant encoder. All heavy contractions run on
// V_WMMA_F32_16X16X4_F32 (native f32 matrix pipe, wave32).
// All register arrays are compile-time indexed (no scratch spills); the one
// genuinely dynamic per-i scalar gather goes through a padded LDS slab.
// Weights are pre-swizzled on device so every WMMA B tile is ONE b64 load.
// ---------------------------------------------------------------------------

typedef float v2f __attribute__((ext_vector_type(2)));
typedef float v8f __attribute__((ext_vector_type(8)));

__device__ __forceinline__ v8f wmma_f32(v2f a, v2f b, v8f c) {
    // D = A(16x4 f32) * B(4x16 f32) + C(16x16 f32)
    return __builtin_amdgcn_wmma_f32_16x16x4_f32(false, a, false, b,
                                                 (short)0, c, false, false);
}

__device__ __forceinline__ v8f zero8() {
    v8f z = {0.f, 0.f, 0.f, 0.f, 0.f, 0.f, 0.f, 0.f};
    return z;
}

// ---------------------------------------------------------------------------
// Weight repack: W[DIN,7,64] -> chunks of 256 floats per (k,s):
//   Wp[(k*S+s)*256 + c2*128 + o*2 + r] = W[4s+2*c2+r, k, o]
// so a lane fetches its B-tile pair {row j, row j+1} with ONE b64 load.
// ---------------------------------------------------------------------------
template <int DIN>
__global__ void pack_w_kernel(const float* __restrict__ W, float* __restrict__ Wp)
{
    constexpr int S = DIN / 4;
    const int idx = blockIdx.x * 256 + threadIdx.x;       // (k, s, c2, o)
    if (idx >= 7 * S * 2 * 64) return;
    const int o  = idx & 63;
    const int c2 = (idx >> 6) & 1;
    const int s  = (idx >> 7) % S;
    const int k  = idx / (128 * S);
    const int i  = 4 * s + 2 * c2;
    float* dst = Wp + ((size_t)(k * S + s) * 2 + c2) * 128 + o * 2;
    dst[0] = W[((size_t)i * 7 + k) * 64 + o];
    dst[1] = W[((size_t)(i + 1) * 7 + k) * 64 + o];
}

// ---------------------------------------------------------------------------
// W3 repack (rectangular, K = 4096): chunk (i,s) at index i*16+s,
// 128 floats per chunk: [c2][o][2] (b64 per lane).
// ---------------------------------------------------------------------------
__global__ void pack_w3_kernel(const float* __restrict__ W3, float* __restrict__ W3q)
{
    const int idx = blockIdx.x * 256 + threadIdx.x;       // (i, s, c2, o)
    if (idx >= 64 * 16 * 2 * 32) return;
    const int o  = idx & 31;
    const int c2 = (idx >> 5) & 1;
    const int s  = (idx >> 6) & 15;
    const int i  = idx >> 10;
    float* dst = W3q + ((size_t)(i * 16 + s) * 2 + c2) * 64 + o * 2;
#pragma unroll
    for (int r2 = 0; r2 < 2; ++r2) {
        const int j = 4 * s + 2 * c2 + r2;
        dst[r2] = W3[((size_t)i * 64 + j) * 32 + o];
    }
}

// ---------------------------------------------------------------------------
// Radial-gated segmented TP over edges (one wave = 32 edges, two M tiles):
//   msg[e,o] = sum_{i,k} x[src[e],i] * basis[e,k] * W[i,k,o] -> atomicAdd hout[dst]
// Loop nest: s (rolled, no register arrays live across it) x k (unrolled, 7).
// ---------------------------------------------------------------------------
template <int DIN>
__global__ __launch_bounds__(128) void edge_tp_kernel(
    const float* __restrict__ xin, const float* __restrict__ pos,
    const int* __restrict__ ei, const float* __restrict__ Wp,
    float* __restrict__ hout, int E)
{
    const int lane = threadIdx.x & 31;
    const int wv   = blockIdx.x * 4 + (threadIdx.x >> 5);
    const long e0  = (long)wv * 32;

    long e = e0 + lane;
    const bool valid = (e < (long)E);
    if (!valid) e = 0;
    const int eSrc = ei[e];
    const int eDst = ei[(long)E + e];

    const float vx = pos[3 * eSrc + 0] - pos[3 * eDst + 0];
    const float vy = pos[3 * eSrc + 1] - pos[3 * eDst + 1];
    const float vz = pos[3 * eSrc + 2] - pos[3 * eDst + 2];
    const float t  = vx * vx + vy * vy + vz * vz + 1e-12f;
    float b[7];
    b[0] = valid ? 1.0f : 0.0f;                // invalid edge -> zero contribution
#pragma unroll
    for (int k = 1; k < 7; ++k) b[k] = b[k - 1] * t;

    const int row  = lane & 15;
    const int coff = (lane < 16) ? 0 : 2;
    const int c2   = coff >> 1;

    const int srcR0 = __shfl(eSrc, row);
    const int srcR1 = __shfl(eSrc, row + 16);
    float bR0[7], bR1[7];
#pragma unroll
    for (int k = 0; k < 7; ++k) {
        bR0[k] = __shfl(b[k], row);
        bR1[k] = __shfl(b[k], row + 16);
    }

    constexpr int S = DIN / 4;
    const float* xp0 = xin + (size_t)srcR0 * DIN + coff;
    const float* xp1 = xin + (size_t)srcR1 * DIN + coff;

    v8f C[2][4];
#pragma unroll
    for (int mt = 0; mt < 2; ++mt)
#pragma unroll
        for (int nt = 0; nt < 4; ++nt) C[mt][nt] = zero8();

#pragma unroll 1
    for (int s = 0; s < S; ++s) {
        const float2 a0 = *(const float2*)(xp0 + 4 * s);
        const float2 a1 = *(const float2*)(xp1 + 4 * s);
        const float* wp = Wp + (size_t)s * 256 + c2 * 128 + row * 2;
#pragma unroll
        for (int k = 0; k < 7; ++k) {
            v2f Aa0, Aa1;
            Aa0.x = a0.x * bR0[k];
            Aa0.y = a0.y * bR0[k];
            Aa1.x = a1.x * bR1[k];
            Aa1.y = a1.y * bR1[k];
            const float* wk = wp + (size_t)k * (S * 256);   // immediate offsets
#pragma unroll
            for (int nt = 0; nt < 4; ++nt) {
                const v2f Bb = *(const v2f*)(wk + nt * 32); // one b64 per B tile
                C[0][nt] = wmma_f32(Aa0, Bb, C[0][nt]);
                C[1][nt] = wmma_f32(Aa1, Bb, C[1][nt]);
            }
        }
    }

#pragma unroll
    for (int mt = 0; mt < 2; ++mt) {
#pragma unroll
        for (int v = 0; v < 8; ++v) {
            const int r = ((lane < 16) ? v : v + 8) + 16 * mt;
            const int d = __shfl(eDst, r);
            float* hp = hout + (size_t)d * 64 + row;
#pragma unroll
            for (int nt = 0; nt < 4; ++nt) atomicAdd(hp + nt * 16, C[mt][nt][v]);
        }
    }
}

// ---------------------------------------------------------------------------
// Node kernel, one wave = 32 nodes (two M tiles):
//   q[n,o]   = sum_{i,j} h[n,i] h[n,j] W3[i,j,o]   (GEMM, K = 4096)
//   out[n,m] = q @ mlp_w + mlp_b                   (GEMM, K = 32)
// A chunks generated on the fly: hi (per-i scalar) from a padded LDS slab
// (row stride 65 -> conflict-free banks), ha[mt][s] constant-indexed.
// ---------------------------------------------------------------------------
__global__ __launch_bounds__(128) void node_quad_mlp_kernel(
    const float* __restrict__ h, const float* __restrict__ W3q,
    const float* __restrict__ mw, const float* __restrict__ mb,
    float* __restrict__ out, int N, int nTiles)
{
    __shared__ __align__(16) float s_mw[4096];          // mlp_w, b64-packed
    __shared__ __align__(16) float s_q[4][16 * 32];     // per-wave transpose slab
    __shared__ __align__(16) float s_h[4][2 * 16 * 65]; // per-wave h rows, padded

    for (int i = threadIdx.x; i < 4096; i += 128) {
        const int o = i >> 7, m = i & 127;
        s_mw[((o >> 2) * 2 + ((o >> 1) & 1)) * 256 + m * 2 + (o & 1)] = mw[i];
    }
    __syncthreads();

    const int lane  = threadIdx.x & 31;
    const int wslot = threadIdx.x >> 5;
    int tile = blockIdx.x * 4 + wslot;
    if (tile >= nTiles) tile = nTiles - 1;              // duplicates write same data
    const int nb   = tile * 32;
    const int row  = lane & 15;
    const int coff = (lane < 16) ? 0 : 2;
    const int c2   = coff >> 1;

    // h rows in WMMA-A layout (registers, constant-indexed only)
    float2 ha[2][16];
#pragma unroll
    for (int mt = 0; mt < 2; ++mt) {
        int node = nb + 16 * mt + row;
        if (node >= N) node = N - 1;
        const float* hp = h + (size_t)node * 64 + coff;
#pragma unroll
        for (int s = 0; s < 16; ++s) ha[mt][s] = *(const float2*)(hp + 4 * s);
    }

    // mirror h rows into padded LDS for the per-i scalar gather
    float* sh = s_h[wslot];
#pragma unroll
    for (int mt = 0; mt < 2; ++mt)
#pragma unroll
        for (int s = 0; s < 16; ++s) {
            float* p = sh + (mt * 16 + row) * 65 + 4 * s + coff;
            p[0] = ha[mt][s].x;
            p[1] = ha[mt][s].y;
        }
    asm volatile("s_wait_dscnt 0x0" ::: "memory");      // LDS in-order per wave

    v8f C[2][2];
#pragma unroll
    for (int mt = 0; mt < 2; ++mt) { C[mt][0] = zero8(); C[mt][1] = zero8(); }

    const float* wp_i = W3q + (size_t)c2 * 64 + row * 2;  // running chunk pointer
#pragma unroll 1
    for (int i = 0; i < 64; ++i) {
        const float hi0 = sh[(0 * 16 + row) * 65 + i];    // ds_load, no bank conflict
        const float hi1 = sh[(1 * 16 + row) * 65 + i];
        __builtin_prefetch(wp_i + 2048, 0, 0);            // next i slab of W3q
#pragma unroll
        for (int s = 0; s < 16; ++s) {
            v2f Aa0, Aa1;
            Aa0.x = hi0 * ha[0][s].x;
            Aa0.y = hi0 * ha[0][s].y;
            Aa1.x = hi1 * ha[1][s].x;
            Aa1.y = hi1 * ha[1][s].y;
#pragma unroll
            for (int nt = 0; nt < 2; ++nt) {
                const v2f Bb = *(const v2f*)(wp_i + s * 128 + nt * 32);
                C[0][nt] = wmma_f32(Aa0, Bb, C[0][nt]);
                C[1][nt] = wmma_f32(Aa1, Bb, C[1][nt]);
            }
        }
        wp_i += 2048;
    }

    // per M-tile: transpose q via LDS, MLP GEMM, bias, store
    float* sq = s_q[wslot];
#pragma unroll 1
    for (int mt = 0; mt < 2; ++mt) {
#pragma unroll
        for (int nt = 0; nt < 2; ++nt)
#pragma unroll
            for (int v = 0; v < 8; ++v) {
                const int r = (lane < 16) ? v : v + 8;
                sq[r * 32 + nt * 16 + row] = C[mt][nt][v];
            }
        asm volatile("s_wait_dscnt 0x0" ::: "memory");

        float2 qa[8];
#pragma unroll
        for (int s = 0; s < 8; ++s)
            qa[s] = *(const float2*)(sq + row * 32 + 4 * s + coff);

        v8f O[8];
#pragma unroll
        for (int nt = 0; nt < 8; ++nt) O[nt] = zero8();

#pragma unroll
        for (int s = 0; s < 8; ++s) {
            v2f Aa;
            Aa.x = qa[s].x;
            Aa.y = qa[s].y;
            const float* wp = s_mw + (s * 2 + c2) * 256 + row * 2;
#pragma unroll
            for (int nt = 0; nt < 8; ++nt) {
                const v2f Bb = *(const v2f*)(wp + nt * 32);
                O[nt] = wmma_f32(Aa, Bb, O[nt]);
            }
        }

#pragma unroll
        for (int nt = 0; nt < 8; ++nt) {
            const float bb = mb[nt * 16 + row];
#pragma unroll
            for (int v = 0; v < 8; ++v) {
                int rn = nb + 16 * mt + ((lane < 16) ? v : v + 8);
                if (rn >= N) rn = N - 1;
                out[(size_t)rn * 128 + nt * 16 + row] = O[nt][v] + bb;
            }
        }
    }
}

// ---------------------------------------------------------------------------
// Elementwise helpers
// ---------------------------------------------------------------------------
__global__ void zero_kernel(float* __restrict__ p, long n)
{
    long i = (long)blockIdx.x * blockDim.x + threadIdx.x;
    if (i < n) p[i] = 0.0f;
}

__global__ void gelu_kernel(float* __restrict__ p, long n)
{
    long i = (long)blockIdx.x * blockDim.x + threadIdx.x;
    if (i < n) {
        const float v = p[i];
        p[i] = 0.5f * v * (1.0f + erff(v * 0.70710678118654752f));
    }
}

// ---------------------------------------------------------------------------
// Launch: pack(W1,W2,W3) + zero -> edgeTP1 -> gelu -> edgeTP2 -> gelu -> node
// ---------------------------------------------------------------------------
extern "C" void kernel_launch(void* const* d_in, const int* in_sizes, int n_in,
                              void* d_out, int out_size, void* d_ws, size_t ws_size,
                              hipStream_t stream)
{
    const float* x   = (const float*)d_in[0];   // [N,32]
    const float* pos = (const float*)d_in[1];   // [N,3]
    const int*   ei  = (const int*)  d_in[2];   // [2,E]
    const float* W1  = (const float*)d_in[3];   // [32,7,64]
    const float* W2  = (const float*)d_in[4];   // [64,7,64]
    const float* W3  = (const float*)d_in[5];   // [64,64,32]
    const float* mw  = (const float*)d_in[6];   // [32,128]
    const float* mb  = (const float*)d_in[7];   // [128]
    float* out = (float*)d_out;

    const int N = in_sizes[1] / 3;
    const int E = in_sizes[2] / 2;

    float* h1  = (float*)d_ws;                  // [N,64]
    float* h2  = h1 + (size_t)N * 64;           // [N,64]
    float* W1p = h2 + (size_t)N * 64;           // 7*8*256   = 14336 floats
    float* W2p = W1p + 7 * 8 * 256;             // 7*16*256  = 28672 floats
    float* W3q = W2p + 7 * 16 * 256;            // 1024*128  = 131072 floats

    pack_w_kernel<32><<<(7 * 8 * 2 * 64 + 255) / 256, 256, 0, stream>>>(W1, W1p);
    pack_w_kernel<64><<<(7 * 16 * 2 * 64 + 255) / 256, 256, 0, stream>>>(W2, W2p);
    pack_w3_kernel<<<(64 * 16 * 2 * 32 + 255) / 256, 256, 0, stream>>>(W3, W3q);

    const long nz = (long)N * 128;              // zero h1 + h2
    zero_kernel<<<(int)((nz + 255) / 256), 256, 0, stream>>>(h1, nz);

    const int eblocks = (E + 127) / 128;        // 4 waves/block, 32 edges/wave
    edge_tp_kernel<32><<<eblocks, 128, 0, stream>>>(x, pos, ei, W1p, h1, E);

    const long nh = (long)N * 64;
    gelu_kernel<<<(int)((nh + 255) / 256), 256, 0, stream>>>(h1, nh);

    edge_tp_kernel<64><<<eblocks, 128, 0, stream>>>(h1, pos, ei, W2p, h2, E);

    gelu_kernel<<<(int)((nh + 255) / 256), 256, 0, stream>>>(h2, nh);

    const int nTiles  = (N + 31) / 32;          // 32 nodes per wave
    const int nblocks = (nTiles + 3) / 4;
    node_quad_mlp_kernel<<<nblocks, 128, 0, stream>>>(h2, W3q, mw, mb, out, N, nTiles);
}